// MODWTLayer_61701500175056
// MI455X (gfx1250) — compile-verified
//
#include <hip/hip_runtime.h>
#include <stdint.h>

// MODWT 5-level, sym4, circular. Fused single pass:
//   read x once (64MB) + write 3 outputs (192MB) ~= 256MB -> ~11us @ 23.3TB/s.
// FMA work offloaded to v_wmma_f32_16x16x4_f32 (Toeplitz-B conv-as-GEMM),
// input staged global->LDS with gfx1250 async-to-LDS DMA.

typedef float v2f __attribute__((ext_vector_type(2)));
typedef float v8f __attribute__((ext_vector_type(8)));

#define NB      16          // batches
#define TT      1048576     // T
#define S_OUT   1984        // outputs per tile
#define CR      2048        // computed range per level = S_OUT + 64 (8 x 256)
#define BUF     2056        // LDS floats per buffer = CR + 8  ([t0-36, t0+S+36))
#define NTILES  529         // ceil(TT / S_OUT)
#define NTHREADS 256

__device__ __forceinline__ v8f wmma_acc(v2f a, v2f b, v8f c) {
  return __builtin_amdgcn_wmma_f32_16x16x4_f32(
      /*neg_a=*/false, a, /*neg_b=*/false, b,
      /*c_mod=*/(short)0, c, /*reuse_a=*/false, /*reuse_b=*/false);
}

// One 256-output unit: D[m][n] = sum_j A[m][j] * B[j][n], K=24 as 6 chained WMMAs.
// A[m][j] = cur[256*g + 16*m + j]  (positions u0+16m-4+j, u0 = t0-32+256g)
__device__ __forceinline__ void conv_unit2(const float* cur, int g, int lane,
                                           const v2f* Blo, const v2f* Bhi,
                                           v8f& lo, v8f& hi, bool do_hi) {
  int m = lane & 15, half = (lane >> 4) & 1;
  int base = 256 * g + 16 * m + 2 * half;
  lo = v8f{};
  hi = v8f{};
#pragma unroll
  for (int c = 0; c < 6; ++c) {
    v2f a = *(const v2f*)(cur + base + 4 * c);   // ds_load_b64, 8B aligned
    lo = wmma_acc(a, Blo[c], lo);
    if (do_hi) hi = wmma_acc(a, Bhi[c], hi);
  }
}

// Scatter D (16x16 f32 C/D layout: lane%16 = N, vgpr r -> M = r + 8*(lane/16))
// into next-level LDS buffer at positions [t0-32, t0+S+32)  (buffer idx 4..4+CR).
__device__ __forceinline__ void store_lds(float* dst, int g, int lane, v8f v) {
  int n = lane & 15, half = (lane >> 4) & 1;
  int base = 4 + 256 * g + 128 * half + n;
#pragma unroll
  for (int r = 0; r < 8; ++r) dst[base + 16 * r] = v[r];
}

// Store D to global output channel, masked to the tile's valid range.
__device__ __forceinline__ void store_out(float* dst, int t0, int g, int lane, v8f v) {
  int n = lane & 15, half = (lane >> 4) & 1;
  int tbase = t0 - 32 + 256 * g + 128 * half + n;
#pragma unroll
  for (int r = 0; r < 8; ++r) {
    int t = tbase + 16 * r;
    if (t >= t0 && t < t0 + S_OUT && t < TT)
      __builtin_nontemporal_store(v[r], dst + t);
  }
}

__global__ void __launch_bounds__(NTHREADS)
modwt_fused_kernel(const float* __restrict__ x, const float* __restrict__ lopass,
                   const float* __restrict__ hipass, float* __restrict__ out) {
  __shared__ float bufA[BUF];
  __shared__ float bufB[BUF];
  __shared__ float wsh[16];   // [0..7]=lowpass, [8..15]=highpass

  const int b    = blockIdx.x / NTILES;
  const int tile = blockIdx.x % NTILES;
  const int t0   = tile * S_OUT;
  const int tid  = threadIdx.x;
  const int lane = tid & 31;
  const int g    = tid >> 5;           // wave id = 256-output unit id, 0..7

  if (tid < 8)            wsh[tid] = lopass[tid];
  else if (tid < 16)      wsh[tid] = hipass[tid - 8];
  __syncthreads();

  // ---- async DMA: global -> LDS, with circular wrap (per-lane gather) ----
  const float* xb = x + (size_t)b * TT;
  for (int i = tid; i < BUF; i += NTHREADS) {
    int gi = t0 - 36 + i;
    if (gi < 0)    gi += TT;
    if (gi >= TT)  gi -= TT;
    unsigned ldsoff = (unsigned)(uintptr_t)(&bufA[i]);
    const float* gp = xb + gi;
    asm volatile("global_load_async_to_lds_b32 %0, %1, off"
                 :: "v"(ldsoff), "v"(gp) : "memory");
  }

  // ---- build Toeplitz B matrices in registers while DMA is in flight ----
  // B layout (4x16): lane%16 = column n; vgpr v holds row K = v + 2*(lane/16).
  const int n = lane & 15, half = (lane >> 4) & 1;
  v2f Blo[6], Bhi[6];
#pragma unroll
  for (int c = 0; c < 6; ++c) {
    int k0 = 4 * c + 2 * half;
    int d0 = k0 - n, d1 = k0 + 1 - n;
    Blo[c].x = (d0 >= 0 && d0 < 8) ? wsh[d0]     : 0.0f;
    Blo[c].y = (d1 >= 0 && d1 < 8) ? wsh[d1]     : 0.0f;
    Bhi[c].x = (d0 >= 0 && d0 < 8) ? wsh[8 + d0] : 0.0f;
    Bhi[c].y = (d1 >= 0 && d1 < 8) ? wsh[8 + d1] : 0.0f;
  }

  asm volatile("s_wait_asynccnt 0" ::: "memory");
  __syncthreads();

  float* cur = bufA;
  float* nxt = bufB;

  // ---- levels 1..3: lowpass only ----
#pragma unroll
  for (int l = 0; l < 3; ++l) {
    v8f lo, hi;
    conv_unit2(cur, g, lane, Blo, Bhi, lo, hi, /*do_hi=*/false);
    store_lds(nxt, g, lane, lo);
    __syncthreads();
    float* t = cur; cur = nxt; nxt = t;
  }

  // ---- level 4: lowpass -> LDS, highpass = h4 -> out channel 0 ----
  {
    v8f lo, hi;
    conv_unit2(cur, g, lane, Blo, Bhi, lo, hi, /*do_hi=*/true);
    store_lds(nxt, g, lane, lo);
    store_out(out + ((size_t)b * 3 + 0) * TT, t0, g, lane, hi);
    __syncthreads();
    float* t = cur; cur = nxt; nxt = t;
  }

  // ---- level 5: highpass = h5 -> channel 1, lowpass = low5 -> channel 2 ----
  {
    v8f lo, hi;
    conv_unit2(cur, g, lane, Blo, Bhi, lo, hi, /*do_hi=*/true);
    store_out(out + ((size_t)b * 3 + 1) * TT, t0, g, lane, hi);
    store_out(out + ((size_t)b * 3 + 2) * TT, t0, g, lane, lo);
  }
}

extern "C" void kernel_launch(void* const* d_in, const int* in_sizes, int n_in,
                              void* d_out, int out_size, void* d_ws, size_t ws_size,
                              hipStream_t stream) {
  const float* x  = (const float*)d_in[0];
  const float* lo = (const float*)d_in[1];
  const float* hi = (const float*)d_in[2];
  float* out = (float*)d_out;

  dim3 grid(NB * NTILES);   // 16 * 529 = 8464 blocks, 8 waves each
  dim3 block(NTHREADS);
  hipLaunchKernelGGL(modwt_fused_kernel, grid, block, 0, stream, x, lo, hi, out);
}